// ModelNew_17514876633330
// MI455X (gfx1250) — compile-verified
//
#include <hip/hip_runtime.h>

typedef __attribute__((ext_vector_type(2))) float v2f;
typedef __attribute__((ext_vector_type(8))) float v8f;

#define NMAT 4096
#define BM 128
#define BN 128
#define BK 32
#define LDA 36    // padded stride for A tile (floats): keeps float2 8B-aligned, conflict-free
#define LDB 136   // padded stride for B tile (floats): 2*136 mod 64 = 16 -> half-waves hit disjoint banks

__global__ __launch_bounds__(256, 2)
void tril_gemm_wmma_f32(const float* __restrict__ A,
                        const float* __restrict__ B,
                        float* __restrict__ C) {
  const int bj = blockIdx.x;   // column block
  const int bi = blockIdx.y;   // row block
  const int tid = threadIdx.x;

  // ---- Upper-triangular blocks: output is strictly zero (and d_out is poisoned) ----
  if (bi < bj) {
    const float4 z = {0.f, 0.f, 0.f, 0.f};
#pragma unroll
    for (int r = 0; r < 16; ++r) {
      int idx  = tid + r * 256;      // 0..4095 float4 slots
      int row  = idx >> 5;           // 32 float4 per 128-col row
      int col4 = (idx & 31) << 2;
      *(float4*)(C + (size_t)(bi * BM + row) * NMAT + bj * BN + col4) = z;
    }
    return;
  }

  __shared__ float As[BM][LDA];  // 128 x 36 floats
  __shared__ float Bs[BK][LDB];  // 32 x 136 floats

  const int lane  = tid & 31;
  const int wave  = tid >> 5;
  const int wm    = wave & 3;    // wave's M slot: 4 waves along M (32 rows each)
  const int wn    = wave >> 2;   // wave's N slot: 2 waves along N (64 cols each)
  const int ln15  = lane & 15;
  const int khalf = lane >> 4;   // 0: K pair (0,1)  1: K pair (2,3)

  v8f acc[2][4];
#pragma unroll
  for (int i = 0; i < 2; ++i)
#pragma unroll
    for (int j = 0; j < 4; ++j)
#pragma unroll
      for (int v = 0; v < 8; ++v) acc[i][j][v] = 0.f;

  // Triangular K window: A[r,k]=0 for k>r, B[k,c]=0 for k<c.
  const int k_begin = bj * BN;
  const int k_end   = (bi + 1) * BM;

  const float* Ablk = A + (size_t)bi * BM * NMAT;  // A rows for this block
  const size_t bcol = (size_t)bj * BN;

  for (int kt = k_begin; kt < k_end; kt += BK) {
    // ---- Stage A tile (128 x 32) into LDS: 4 float4 per thread ----
#pragma unroll
    for (int r = 0; r < 4; ++r) {
      int idx = tid + r * 256;       // 0..1023
      int row = idx >> 3;            // 8 float4 per 32-col row
      int col = (idx & 7) << 2;
      const float4 v = *(const float4*)(Ablk + (size_t)row * NMAT + kt + col);
      *(float4*)(&As[row][col]) = v;
      if (kt + BK < k_end)
        __builtin_prefetch(Ablk + (size_t)row * NMAT + kt + BK + col, 0, 3);
    }
    // ---- Stage B tile (32 x 128) into LDS: 4 float4 per thread ----
#pragma unroll
    for (int r = 0; r < 4; ++r) {
      int idx = tid + r * 256;
      int row = idx >> 5;            // 32 float4 per 128-col row
      int col = (idx & 31) << 2;
      const float4 v = *(const float4*)(B + (size_t)(kt + row) * NMAT + bcol + col);
      *(float4*)(&Bs[row][col]) = v;
      if (kt + BK < k_end)
        __builtin_prefetch(B + (size_t)(kt + BK + row) * NMAT + bcol + col, 0, 3);
    }
    __syncthreads();

    // ---- 8 WMMA K-steps of K=4 each ----
#pragma unroll
    for (int kk = 0; kk < BK; kk += 4) {
      const int kb = kk + khalf * 2;

      // A frags: lane holds M = ln15 (row of its 16x4 tile), K pair (kb, kb+1)
      v2f afr[2];
#pragma unroll
      for (int i = 0; i < 2; ++i) {
        const int m = wm * 32 + i * 16 + ln15;
        afr[i] = *(const v2f*)(&As[m][kb]);   // 8B-aligned: (m*36 + kb) even
      }
      // B frags: lane holds N = ln15, K pair (kb, kb+1)
      v2f bfr[4];
#pragma unroll
      for (int j = 0; j < 4; ++j) {
        const int n = wn * 64 + j * 16 + ln15;
        bfr[j].x = Bs[kb][n];
        bfr[j].y = Bs[kb + 1][n];
      }

#pragma unroll
      for (int i = 0; i < 2; ++i)
#pragma unroll
        for (int j = 0; j < 4; ++j)
          acc[i][j] = __builtin_amdgcn_wmma_f32_16x16x4_f32(
              false, afr[i], false, bfr[j], (short)0, acc[i][j], false, false);
    }
    __syncthreads();
  }

  // ---- Store: C/D layout = VGPR v -> M=v (lanes 0-15) / M=v+8 (lanes 16-31), N=lane&15 ----
  const bool diag = (bi == bj);
#pragma unroll
  for (int i = 0; i < 2; ++i) {
#pragma unroll
    for (int j = 0; j < 4; ++j) {
      const int gcol  = bj * BN + wn * 64 + j * 16 + ln15;
      const int grow0 = bi * BM + wm * 32 + i * 16 + khalf * 8;
#pragma unroll
      for (int v = 0; v < 8; ++v) {
        const int grow = grow0 + v;
        float val = acc[i][j][v];
        if (diag && (gcol > grow)) val = 0.f;   // tril mask on diagonal blocks
        C[(size_t)grow * NMAT + gcol] = val;
      }
    }
  }
}

extern "C" void kernel_launch(void* const* d_in, const int* in_sizes, int n_in,
                              void* d_out, int out_size, void* d_ws, size_t ws_size,
                              hipStream_t stream) {
  const float* A = (const float*)d_in[0];
  const float* B = (const float*)d_in[1];
  float* C = (float*)d_out;
  (void)in_sizes; (void)n_in; (void)out_size; (void)d_ws; (void)ws_size;

  dim3 grid(NMAT / BN, NMAT / BM);   // 32 x 32 block tiles
  tril_gemm_wmma_f32<<<grid, 256, 0, stream>>>(A, B, C);
}